// SafetyLayerAgg_40226663694894
// MI455X (gfx1250) — compile-verified
//
#include <hip/hip_runtime.h>

// Elementwise clip: out[i] = min(max(yhat[i], obs[2i]), obs[2i+1])
// Memory-bound streaming kernel: 16 B traffic / element, 2 FLOP / element.
// Strategy: B128 vector loads/stores with non-temporal cache hints (no reuse,
// keep the 192MB L2 clean), 4 elements per lane, wave32 blocks of 256.

typedef __attribute__((ext_vector_type(4))) float v4f;

__global__ __launch_bounds__(256) void
SafetyLayer_clip_v4(const float* __restrict__ yhat,
                    const float* __restrict__ obs,
                    float* __restrict__ out,
                    int n4, int n) {
    int i = blockIdx.x * blockDim.x + threadIdx.x;
    if (i < n4) {
        const v4f* y4 = (const v4f*)yhat;
        const v4f* o4 = (const v4f*)obs;   // obs is [lower,upper] interleaved
        v4f*       r4 = (v4f*)out;

        // 1x B128 for yhat, 2x B128 for the 8 obs floats covering these 4 rows
        v4f y  = __builtin_nontemporal_load(y4 + i);
        v4f o0 = __builtin_nontemporal_load(o4 + 2 * i);
        v4f o1 = __builtin_nontemporal_load(o4 + 2 * i + 1);

        v4f r;
        r.x = __builtin_fminf(__builtin_fmaxf(y.x, o0.x), o0.y);
        r.y = __builtin_fminf(__builtin_fmaxf(y.y, o0.z), o0.w);
        r.z = __builtin_fminf(__builtin_fmaxf(y.z, o1.x), o1.y);
        r.w = __builtin_fminf(__builtin_fmaxf(y.w, o1.z), o1.w);

        __builtin_nontemporal_store(r, r4 + i);
    }

    // Scalar tail for n % 4 != 0 (dead at B = 16M, kept for generality).
    if (blockIdx.x == 0 && threadIdx.x == 0) {
        for (int j = n4 * 4; j < n; ++j) {
            float lo = obs[2 * j];
            float hi = obs[2 * j + 1];
            out[j] = __builtin_fminf(__builtin_fmaxf(yhat[j], lo), hi);
        }
    }
}

extern "C" void kernel_launch(void* const* d_in, const int* in_sizes, int n_in,
                              void* d_out, int out_size, void* d_ws, size_t ws_size,
                              hipStream_t stream) {
    const float* yhat = (const float*)d_in[0];   // (B,1) float32
    const float* obs  = (const float*)d_in[1];   // (B,2) float32, [lower,upper]
    float*       out  = (float*)d_out;           // (B,1) float32

    int n  = out_size;       // B elements
    int n4 = n >> 2;         // 4 elements per thread

    int block = 256;
    int grid  = (n4 + block - 1) / block;
    if (grid < 1) grid = 1;

    SafetyLayer_clip_v4<<<grid, block, 0, stream>>>(yhat, obs, out, n4, n);
}